// SparseTransformerHead_49357764166121
// MI455X (gfx1250) — compile-verified
//
#include <hip/hip_runtime.h>
#include <math.h>

#define Dm    512
#define Hh    8
#define Ll    4
#define Bb    8
#define Tt    511
#define TcL   512
#define FFd   256
#define HEADH 32
#define HDd   64
#define NEGV  (-1e9f)

// GEMM tiling
#define BLK_M 32
#define BLK_N 128
#define CH_K  32
#define ASTR  36     // padded A row stride (floats): conflict-free, 16B aligned
#define BSTR  132    // padded B row stride (floats): kills half-wave bank conflict

typedef __attribute__((ext_vector_type(2))) float v2f;
typedef __attribute__((ext_vector_type(8))) float v8f;
typedef __attribute__((ext_vector_type(4))) unsigned int u32x4;
typedef __attribute__((ext_vector_type(8))) int   i32x8;
typedef __attribute__((ext_vector_type(4))) int   i32x4;

#if __has_builtin(__builtin_amdgcn_tensor_load_to_lds) && __has_builtin(__builtin_amdgcn_s_wait_tensorcnt)
#define STX_TDM 1
#else
#define STX_TDM 0
#endif

// ---------------------------------------------------------------------------
// TDM 2D tile load: global (row-major, stride0 elems/row) -> LDS with
// descriptor padding (padIntervalCode: dwords=2<<code, padAmountCode: dw=code+1)
// ---------------------------------------------------------------------------
__device__ __forceinline__ void stx_tdm_load_tile(
    unsigned ldsAddr, const float* gsrc,
    unsigned tensor_d0, unsigned tensor_d1, unsigned stride0,
    unsigned tile_w, unsigned tile_h,
    unsigned padIntervalCode, unsigned padAmountCode)
{
#if STX_TDM
    const unsigned long long ga = (unsigned long long)gsrc;
    u32x4 g0;
    g0[0] = 1u;                                                  // count=1, user D#
    g0[1] = ldsAddr;                                             // lds_addr
    g0[2] = (unsigned)(ga & 0xffffffffull);                      // global_addr[31:0]
    g0[3] = (unsigned)((ga >> 32) & 0x01ffffffull) | (2u << 30); // addr[56:32] | type=2
    i32x8 g1;
    // data_size=4B | pad_enable | pad_interval | pad_amount
    g1[0] = (int)((2u << 16) | (1u << 20) | (padIntervalCode << 22) | (padAmountCode << 25));
    g1[1] = (int)((tensor_d0 & 0xffffu) << 16);                                  // dim0[15:0]
    g1[2] = (int)(((tensor_d0 >> 16) & 0xffffu) | ((tensor_d1 & 0xffffu) << 16));// dim0[31:16] dim1[15:0]
    g1[3] = (int)(((tensor_d1 >> 16) & 0xffffu) | ((tile_w & 0xffffu) << 16));   // dim1[31:16] tile_dim0
    g1[4] = (int)(tile_h & 0xffffu);                                             // tile_dim1, tile_dim2=0
    g1[5] = (int)stride0;                                                        // stride0[31:0]
    g1[6] = 0;                                                                   // stride0[47:32], d1stride lo
    g1[7] = 0;
    const i32x4 z4 = {0, 0, 0, 0};
    const i32x8 z8 = {0, 0, 0, 0, 0, 0, 0, 0};
    __builtin_amdgcn_tensor_load_to_lds(g0, g1, z4, z4, z8, 0);
#else
    (void)ldsAddr; (void)gsrc; (void)tensor_d0; (void)tensor_d1; (void)stride0;
    (void)tile_w; (void)tile_h; (void)padIntervalCode; (void)padAmountCode;
#endif
}

// ---------------------------------------------------------------------------
// Embedding + positional + CLS prepend + pad-mask build.
// ---------------------------------------------------------------------------
__global__ __launch_bounds__(256)
void stx_embed_kernel(const int* __restrict__ x, const float* __restrict__ emb,
                      const float* __restrict__ pos, const float* __restrict__ cls,
                      float* __restrict__ h, float* __restrict__ pad)
{
    const int idx = blockIdx.x;            // b*Tc + t
    const int t = idx % TcL, b = idx / TcL;
    const int tid = threadIdx.x;
    if (t == 0) {
        h[(size_t)idx * Dm + tid]       = cls[tid];
        h[(size_t)idx * Dm + tid + 256] = cls[tid + 256];
        if (tid == 0) pad[idx] = 0.0f;
    } else {
        const int tok = x[b * Tt + (t - 1)];
#pragma unroll
        for (int r = 0; r < 2; ++r) {
            const int d = tid + r * 256;
            const float e = (tok != 0) ? emb[(size_t)tok * Dm + d] : 0.0f;
            h[(size_t)idx * Dm + d] = e + pos[(size_t)(t - 1) * Dm + d];
        }
        if (tid == 0) pad[idx] = (tok == 0) ? 1.0f : 0.0f;
    }
}

// ---------------------------------------------------------------------------
// LayerNorm over D=512 per row; optional post-add: y = LN(x)*g+b [+ add]
// ---------------------------------------------------------------------------
__global__ __launch_bounds__(256)
void stx_ln_kernel(const float* __restrict__ x, const float* __restrict__ g,
                   const float* __restrict__ b, const float* __restrict__ add,
                   float* __restrict__ y)
{
    __shared__ float red[256];
    const int row = blockIdx.x;
    const int tid = threadIdx.x;
    const float* xr = x + (size_t)row * Dm;
    const float v0 = xr[tid], v1 = xr[tid + 256];

    red[tid] = v0 + v1;
    __syncthreads();
    for (int s = 128; s > 0; s >>= 1) { if (tid < s) red[tid] += red[tid + s]; __syncthreads(); }
    const float mean = red[0] * (1.0f / (float)Dm);
    __syncthreads();

    const float d0 = v0 - mean, d1 = v1 - mean;
    red[tid] = d0 * d0 + d1 * d1;
    __syncthreads();
    for (int s = 128; s > 0; s >>= 1) { if (tid < s) red[tid] += red[tid + s]; __syncthreads(); }
    const float rstd = rsqrtf(red[0] * (1.0f / (float)Dm) + 1e-5f);

    float o0 = d0 * rstd * g[tid]       + b[tid];
    float o1 = d1 * rstd * g[tid + 256] + b[tid + 256];
    if (add) {
        o0 += add[(size_t)row * Dm + tid];
        o1 += add[(size_t)row * Dm + tid + 256];
    }
    y[(size_t)row * Dm + tid]       = o0;
    y[(size_t)row * Dm + tid + 256] = o1;
}

// ---------------------------------------------------------------------------
// Fused GEMM: Y[M,N] = act(X[M,K] @ W[K,N] + bias) [+ res]
// act: 0=none, 1=exact GELU, 2=relu.
// Block tile 32x128, K-chunk 32; 8 waves = 2(M) x 4(N), two 16x16 f32
// accumulators per wave. Both A and B tiles are moved by the Tensor Data
// Mover into double-buffered LDS (descriptor padding -> strides 36 / 132,
// bank-conflict free). TENSORcnt is in-order, so the loop keeps the NEXT
// chunk's pair of DMAs in flight while computing the current one and waits
// s_wait_tensorcnt <= 2. Inner pipeline has zero per-thread global loads.
// ---------------------------------------------------------------------------
__global__ __launch_bounds__(256)
void stx_gemm_wmma_kernel(const float* __restrict__ X, const float* __restrict__ W,
                          const float* __restrict__ bias, const float* __restrict__ res,
                          float* __restrict__ Y, int M, int N, int K, int act)
{
    __shared__ float A_s[2][BLK_M * ASTR];
    __shared__ float B_s[2][CH_K * BSTR];

    const int tid    = threadIdx.x;
    const int lane   = tid & 31;
    const int waveId = tid >> 5;
    const int half   = lane >> 4;      // 0: K=+0,+1   1: K=+2,+3
    const int l16    = lane & 15;
    const int nBlkN  = N / BLK_N;
    const int row0   = (blockIdx.x / nBlkN) * BLK_M;
    const int col0   = (blockIdx.x % nBlkN) * BLK_N;
    const int wm     = waveId >> 2;    // 0..1
    const int wn     = waveId & 3;     // 0..3
    const int nch    = K / CH_K;

    const float* Abase = X + (size_t)row0 * K;   // tile row origin
    const float* Bbase = W + col0;               // tile col origin

#if STX_TDM
    // ---- prologue: chunk 0 DMAs in flight ----
    if (waveId == 0) {
        stx_tdm_load_tile((unsigned)(unsigned long long)&A_s[0][0], Abase,
                          (unsigned)K, (unsigned)M, (unsigned)K, CH_K, BLK_M, 4u, 3u);
        stx_tdm_load_tile((unsigned)(unsigned long long)&B_s[0][0], Bbase,
                          (unsigned)N, (unsigned)K, (unsigned)N, BLK_N, CH_K, 6u, 3u);
    }
#endif

    v8f acc0 = {}, acc1 = {};
    for (int i = 0; i < nch; ++i) {
        const int cur = i & 1;
#if STX_TDM
        if (waveId == 0) {
            if (i + 1 < nch) {
                const int nk0 = (i + 1) * CH_K;
                stx_tdm_load_tile((unsigned)(unsigned long long)&A_s[cur ^ 1][0], Abase + nk0,
                                  (unsigned)K, (unsigned)M, (unsigned)K, CH_K, BLK_M, 4u, 3u);
                stx_tdm_load_tile((unsigned)(unsigned long long)&B_s[cur ^ 1][0],
                                  Bbase + (size_t)nk0 * N,
                                  (unsigned)N, (unsigned)K, (unsigned)N, BLK_N, CH_K, 6u, 3u);
                __builtin_amdgcn_s_wait_tensorcnt((unsigned short)2);  // chunk i done; i+1 in flight
            } else {
                __builtin_amdgcn_s_wait_tensorcnt((unsigned short)0);  // final pair done
            }
        }
#else
        {   // manual staging fallback (coalesced float4)
            const int k0 = i * CH_K;
            const int am = tid >> 3;           // 0..31
            const int ak = (tid & 7) * 4;      // 0..28
            const float4 av = *(const float4*)(Abase + (size_t)am * K + k0 + ak);
            A_s[cur][am * ASTR + ak + 0] = av.x;
            A_s[cur][am * ASTR + ak + 1] = av.y;
            A_s[cur][am * ASTR + ak + 2] = av.z;
            A_s[cur][am * ASTR + ak + 3] = av.w;
            const int bk = tid >> 5;           // 0..7
            const int bn = (tid & 31) * 4;     // 0..124
            for (int r = 0; r < 4; ++r) {
                const int kk = bk + r * 8;
                *(float4*)(&B_s[cur][kk * BSTR + bn]) =
                    *(const float4*)(Bbase + (size_t)(k0 + kk) * N + bn);
            }
        }
#endif
        __syncthreads();   // chunk i tiles visible to all waves

        // ---- compute: 8 k-steps x 2 WMMA (EXEC all-ones here) ----
#pragma unroll
        for (int kk = 0; kk < CH_K; kk += 4) {
            v2f a, b0, b1;
            const int ar = (wm * 16 + l16) * ASTR + kk + 2 * half;
            a.x = A_s[cur][ar];
            a.y = A_s[cur][ar + 1];
            const int br = (kk + 2 * half) * BSTR + wn * 32 + l16;
            b0.x = B_s[cur][br];
            b0.y = B_s[cur][br + BSTR];
            b1.x = B_s[cur][br + 16];
            b1.y = B_s[cur][br + BSTR + 16];
            acc0 = __builtin_amdgcn_wmma_f32_16x16x4_f32(false, a, false, b0, (short)0, acc0, false, false);
            acc1 = __builtin_amdgcn_wmma_f32_16x16x4_f32(false, a, false, b1, (short)0, acc1, false, false);
        }
        __syncthreads();   // all reads of buf[cur] done before it is re-targeted
    }

    // ---- epilogue: bias / activation / residual, fused ----
    const int colA = col0 + wn * 32 + l16;
    const int colB = colA + 16;
    const float bvA = bias ? bias[colA] : 0.0f;
    const float bvB = bias ? bias[colB] : 0.0f;
#pragma unroll
    for (int i = 0; i < 8; ++i) {
        const int row = row0 + wm * 16 + i + 8 * half;   // C/D layout: VGPR i -> M=i / M=8+i
        float vA = acc0[i] + bvA;
        float vB = acc1[i] + bvB;
        if (act == 1) {
            vA = 0.5f * vA * (1.0f + erff(vA * 0.70710678118654752f));
            vB = 0.5f * vB * (1.0f + erff(vB * 0.70710678118654752f));
        } else if (act == 2) {
            vA = fmaxf(vA, 0.0f);
            vB = fmaxf(vB, 0.0f);
        }
        if (res) {
            vA += res[(size_t)row * N + colA];
            vB += res[(size_t)row * N + colB];
        }
        Y[(size_t)row * N + colA] = vA;
        Y[(size_t)row * N + colB] = vB;
    }
}

// ---------------------------------------------------------------------------
// Entmax-1.5 attention, one query row per block (grid = B*H*Tc, 256 threads).
// scores = q.k/8, key-pad mask, x = s/2 - max; tau via 30-step bisection on
// f(tau)=sum((x-tau)_+^2)=1; p = (x-tau)_+^2 renormalized; o = p @ V.
// ---------------------------------------------------------------------------
__global__ __launch_bounds__(256)
void stx_attn_entmax_kernel(const float* __restrict__ Q, const float* __restrict__ Kb,
                            const float* __restrict__ Vb, const float* __restrict__ pad,
                            float* __restrict__ O)
{
    __shared__ float qs[HDd];
    __shared__ float xs[TcL];
    __shared__ float red[256];
    const int idx = blockIdx.x;
    const int tq  = idx % TcL;
    const int hh  = (idx / TcL) % Hh;
    const int bi  = idx / (TcL * Hh);
    const int tid = threadIdx.x;

    if (tid < HDd) qs[tid] = Q[((size_t)(bi * TcL + tq)) * Dm + hh * HDd + tid];
    __syncthreads();

    float xv[2];
#pragma unroll
    for (int r = 0; r < 2; ++r) {
        const int j = tid + r * 256;
        const float* kr = Kb + ((size_t)(bi * TcL + j)) * Dm + hh * HDd;
        float s = 0.0f;
#pragma unroll 8
        for (int d = 0; d < HDd; ++d) s += qs[d] * kr[d];
        s *= 0.125f;                               // / sqrt(hd=64)
        if (pad[bi * TcL + j] != 0.0f) s = NEGV;   // key padding mask
        xv[r] = 0.5f * s;                          // entmax alpha=1.5 pre-scale
        xs[j] = xv[r];
    }

    red[tid] = fmaxf(xv[0], xv[1]);
    __syncthreads();
    for (int s = 128; s > 0; s >>= 1) { if (tid < s) red[tid] = fmaxf(red[tid], red[tid + s]); __syncthreads(); }
    const float xm = red[0];
    __syncthreads();

    float lo = -1.0f, hi = 0.0f;                   // f(-1) >= 1, f(0) <= 1
    for (int it = 0; it < 30; ++it) {
        const float mid = 0.5f * (lo + hi);
        float s0 = 0.0f;
#pragma unroll
        for (int r = 0; r < 2; ++r) {
            const float t = xv[r] - xm - mid;
            if (t > 0.0f) s0 += t * t;
        }
        red[tid] = s0;
        __syncthreads();
        for (int s = 128; s > 0; s >>= 1) { if (tid < s) red[tid] += red[tid + s]; __syncthreads(); }
        const float f = red[0];
        __syncthreads();
        if (f >= 1.0f) lo = mid; else hi = mid;    // uniform across block
    }
    const float tau = 0.5f * (lo + hi);

    float pv[2], z0 = 0.0f;
#pragma unroll
    for (int r = 0; r < 2; ++r) {
        const float t = xv[r] - xm - tau;
        pv[r] = (t > 0.0f) ? t * t : 0.0f;
        z0 += pv[r];
    }
    red[tid] = z0;
    __syncthreads();
    for (int s = 128; s > 0; s >>= 1) { if (tid < s) red[tid] += red[tid + s]; __syncthreads(); }
    const float inv = 1.0f / red[0];
    __syncthreads();
    xs[tid]       = pv[0] * inv;
    xs[tid + 256] = pv[1] * inv;
    __syncthreads();

    const int d  = tid & 63;
    const int qt = tid >> 6;
    float accum = 0.0f;
    for (int j = qt * 128; j < qt * 128 + 128; ++j)
        accum += xs[j] * Vb[((size_t)(bi * TcL + j)) * Dm + hh * HDd + d];
    red[tid] = accum;
    __syncthreads();
    if (tid < 64) {
        const float o = red[tid] + red[tid + 64] + red[tid + 128] + red[tid + 192];
        O[((size_t)(bi * TcL + tq)) * Dm + hh * HDd + tid] = o;
    }
}

// ---------------------------------------------------------------------------
// Classifier head: z = relu(h[:,0] @ h1_w + h1_b); out = z @ h2_w + h2_b.
// ---------------------------------------------------------------------------
__global__ __launch_bounds__(256)
void stx_head_kernel(const float* __restrict__ h, const float* __restrict__ w1,
                     const float* __restrict__ b1, const float* __restrict__ w2,
                     const float* __restrict__ b2, float* __restrict__ out)
{
    __shared__ float zs[256];
    const int tid = threadIdx.x;
    const int b = tid >> 5;
    const int j = tid & 31;
    const float* hr = h + (size_t)(b * TcL) * Dm;   // CLS row of batch b
    float a = b1[j];
    for (int i = 0; i < Dm; ++i) a += hr[i] * w1[i * HEADH + j];
    zs[tid] = fmaxf(a, 0.0f);
    __syncthreads();
    if (tid < Bb) {
        const float* z = zs + tid * HEADH;
        float o = b2[0];
        for (int j2 = 0; j2 < HEADH; ++j2) o += z[j2] * w2[j2];
        out[tid] = o;
    }
}

// ---------------------------------------------------------------------------
extern "C" void kernel_launch(void* const* d_in, const int* in_sizes, int n_in,
                              void* d_out, int out_size, void* d_ws, size_t ws_size,
                              hipStream_t stream)
{
    (void)in_sizes; (void)n_in; (void)out_size; (void)ws_size;
    const int*   x     = (const int*)  d_in[0];
    const float* emb   = (const float*)d_in[1];
    const float* pos   = (const float*)d_in[2];
    const float* cls   = (const float*)d_in[3];
    const float* Wq    = (const float*)d_in[4];
    const float* Wk    = (const float*)d_in[5];
    const float* Wv    = (const float*)d_in[6];
    const float* Wo    = (const float*)d_in[7];
    const float* an_g  = (const float*)d_in[8];
    const float* an_b  = (const float*)d_in[9];
    const float* ln1_g = (const float*)d_in[10];
    const float* ln1_b = (const float*)d_in[11];
    const float* ln2_g = (const float*)d_in[12];
    const float* ln2_b = (const float*)d_in[13];
    const float* f1w   = (const float*)d_in[14];
    const float* f1b   = (const float*)d_in[15];
    const float* f2w   = (const float*)d_in[16];
    const float* f2b   = (const float*)d_in[17];
    const float* h1w   = (const float*)d_in[18];
    const float* h1b   = (const float*)d_in[19];
    const float* h2w   = (const float*)d_in[20];
    const float* h2b   = (const float*)d_in[21];

    float* ws  = (float*)d_ws;                 // all scratch carved from d_ws
    float* hb  = ws;                           // [4096,512]
    float* xin = ws + 2097152;                 // [4096,512]
    float* qb  = ws + 4194304;                 // [4096,512]
    float* kb  = ws + 6291456;                 // [4096,512]
    float* vb  = ws + 8388608;                 // [4096,512]
    float* att = ws + 10485760;                // [4096,512]
    float* tmp = ws + 12582912;                // [4096,512]
    float* mid = ws + 14680064;                // [4096,256]
    float* pad = ws + 15728640;                // [4096]

    const int Mrows    = Bb * TcL;                              // 4096
    const int blocksDD = (Mrows / BLK_M) * (Dm  / BLK_N);       // 512  (N=512)
    const int blocksDF = (Mrows / BLK_M) * (FFd / BLK_N);       // 256  (N=256)

    stx_embed_kernel<<<Mrows, 256, 0, stream>>>(x, emb, pos, cls, hb, pad);

    for (int l = 0; l < Ll; ++l) {
        const float* wq = Wq + (size_t)l * Dm * Dm;
        const float* wk = Wk + (size_t)l * Dm * Dm;
        const float* wv = Wv + (size_t)l * Dm * Dm;
        const float* wo = Wo + (size_t)l * Dm * Dm;

        stx_ln_kernel<<<Mrows, 256, 0, stream>>>(hb, ln1_g + l * Dm, ln1_b + l * Dm, nullptr, xin);
        stx_gemm_wmma_kernel<<<blocksDD, 256, 0, stream>>>(xin, wq, nullptr, nullptr, qb, Mrows, Dm, Dm, 0);
        stx_gemm_wmma_kernel<<<blocksDD, 256, 0, stream>>>(xin, wk, nullptr, nullptr, kb, Mrows, Dm, Dm, 0);
        stx_gemm_wmma_kernel<<<blocksDD, 256, 0, stream>>>(xin, wv, nullptr, nullptr, vb, Mrows, Dm, Dm, 0);
        stx_attn_entmax_kernel<<<Bb * Hh * TcL, 256, 0, stream>>>(qb, kb, vb, pad, att);
        stx_gemm_wmma_kernel<<<blocksDD, 256, 0, stream>>>(att, wo, nullptr, xin, tmp, Mrows, Dm, Dm, 0);
        stx_ln_kernel<<<Mrows, 256, 0, stream>>>(tmp, an_g + l * Dm, an_b + l * Dm, hb, hb);
        stx_ln_kernel<<<Mrows, 256, 0, stream>>>(hb, ln2_g + l * Dm, ln2_b + l * Dm, nullptr, xin);
        stx_gemm_wmma_kernel<<<blocksDF, 256, 0, stream>>>(xin, f1w + (size_t)l * Dm * FFd,
                                                           f1b + l * FFd, nullptr, mid, Mrows, FFd, Dm, 1);
        stx_gemm_wmma_kernel<<<blocksDD, 256, 0, stream>>>(mid, f2w + (size_t)l * FFd * Dm,
                                                           f2b + l * Dm, hb, hb, Mrows, Dm, FFd, 0);
    }

    stx_head_kernel<<<1, 256, 0, stream>>>(hb, h1w, h1b, h2w, h2b, (float*)d_out);
}